// Net_25503515804348
// MI455X (gfx1250) — compile-verified
//
#include <hip/hip_runtime.h>
#include <hip/hip_bf16.h>

typedef __attribute__((ext_vector_type(2))) float v2f;
typedef __attribute__((ext_vector_type(8))) float v8f;

// ---------------------------------------------------------------------------
// Degree / incoming-count accumulation:  deg over source (row), cnt over target
// ---------------------------------------------------------------------------
__global__ void degree_kernel(const int* __restrict__ ei, float* __restrict__ deg,
                              float* __restrict__ cnt, int E) {
    int e = blockIdx.x * blockDim.x + threadIdx.x;
    if (e >= E) return;
    atomicAdd(&deg[ei[e]], 1.0f);        // row = ei[0:E]
    atomicAdd(&cnt[ei[E + e]], 1.0f);    // col = ei[E:2E]
}

// deg_inv_sqrt = deg > 0 ? rsqrt(max(deg,1)) : 0
__global__ void dis_kernel(const float* __restrict__ deg, float* __restrict__ dis, int N) {
    int i = blockIdx.x * blockDim.x + threadIdx.x;
    if (i >= N) return;
    float d = deg[i];
    dis[i] = (d > 0.0f) ? rsqrtf(fmaxf(d, 1.0f)) : 0.0f;
}

// ---------------------------------------------------------------------------
// Edge message kernel using V_WMMA_F32_16X16X4_F32.
// One wave = 16 edges.  A[16 x K] = norm * concat(feat[row], edge_attr),
// B[K x 16] = Wn, accumulated over ceil(K/4) K=4 WMMA steps.
// A layout: lane m (m=lane&15) holds edge m; low lanes supply K pair {k0,k0+1},
//           high lanes {k0+2,k0+3} (VGPR0/VGPR1 of the v2f).
// B layout: VGPR0 = row k0 (low lanes) / k0+2 (high lanes), col = lane&15.
// D layout: lane holds column n=lane&15, rows r + 8*(lane>=16), r=0..7.
// All per-lane selection is branchless (cndmask on addresses/values); every
// global load is unconditional with an in-bounds address for all lanes, so
// EXEC stays all-1s end-to-end (required by WMMA) and no saveexec is emitted.
// Out-of-range edges get nrm=0 => zero D rows => unconditional atomicAdd(0) no-op.
// ---------------------------------------------------------------------------
template <int F_IN>
__global__ void edge_msg_wmma(const int* __restrict__ ei,
                              const float* __restrict__ feat,   // [N, F_IN]
                              const float* __restrict__ eattr,  // [E, 2]
                              const float* __restrict__ dis,    // [N]
                              const float* __restrict__ Wn,     // [F_IN+2, 16]
                              float* __restrict__ aggr,         // [N, 16] (sum)
                              int E) {
    const int lane = threadIdx.x & 31;
    const int wave = threadIdx.x >> 5;
    const int tile = blockIdx.x * (blockDim.x >> 5) + wave;
    const int base = tile * 16;
    if (base >= E) return;                 // wave-uniform: EXEC stays all-1s

    const int m    = lane & 15;            // edge slot this lane feeds into A
    const int hi   = lane >> 4;            // 0 = low half (K0,K1), 1 = high (K2,K3)
    const int e    = base + m;
    const bool vld = (e < E);
    const int es   = vld ? e : base;       // safe index, keeps EXEC full

    const int   r   = ei[es];
    const int   c   = ei[E + es];
    const float nrm = vld ? dis[r] * dis[c] : 0.0f;  // zero row => zero output
    const float sel = hi ? 0.0f : 1.0f;    // kills K-slots beyond K_TOT on high lanes

    v8f acc = {};

    if constexpr (F_IN == 16) {
        // K_TOT = 18, 5 chunks. Chunks 0..3: pure node features, contiguous pairs.
        const float2* frow = (const float2*)(feat + r * 16);
#pragma unroll
        for (int ch = 0; ch < 4; ++ch) {
            const int k0 = ch * 4 + hi * 2;          // even, < 16
            float2 av = frow[k0 >> 1];               // global_load_b64
            v2f A = { av.x * nrm, av.y * nrm };
            v2f B = { Wn[k0 * 16 + m], Wn[(k0 + 1) * 16 + m] };
            acc = __builtin_amdgcn_wmma_f32_16x16x4_f32(false, A, false, B,
                                                        (short)0, acc, false, false);
        }
        // Chunk 4: low lanes k={16,17} = edge_attr, high lanes k={18,19} = 0.
        {
            float2 ev = *(const float2*)(eattr + es * 2);   // valid for all lanes
            float  w0 = Wn[16 * 16 + m];                    // valid for all lanes
            float  w1 = Wn[17 * 16 + m];                    // valid for all lanes
            v2f A = { ev.x * nrm * sel, ev.y * nrm * sel };
            v2f B = { w0 * sel, w1 * sel };
            acc = __builtin_amdgcn_wmma_f32_16x16x4_f32(false, A, false, B,
                                                        (short)0, acc, false, false);
        }
    } else {
        // F_IN == 3, K_TOT = 5, 2 chunks.
        // Chunk 0: low lanes k={0,1} = feat; high lanes k={2,3} = {feat[2], eattr[0]}.
        {
            const float* p0 = feat + r * 3 + (hi ? 2 : 0);   // branchless addr select
            const float* p1 = hi ? (eattr + es * 2) : (feat + r * 3 + 1);
            v2f A = { p0[0] * nrm, p1[0] * nrm };
            v2f B = { Wn[(hi ? 2 : 0) * 16 + m], Wn[(hi ? 3 : 1) * 16 + m] };
            acc = __builtin_amdgcn_wmma_f32_16x16x4_f32(false, A, false, B,
                                                        (short)0, acc, false, false);
        }
        // Chunk 1: low lanes k={4,5} = {eattr[1], 0}; high lanes k={6,7} = 0.
        {
            float ea1 = eattr[es * 2 + 1];                   // valid for all lanes
            float w4  = Wn[4 * 16 + m];                      // valid for all lanes
            v2f A = { ea1 * nrm * sel, 0.0f };
            v2f B = { w4 * sel, 0.0f };
            acc = __builtin_amdgcn_wmma_f32_16x16x4_f32(false, A, false, B,
                                                        (short)0, acc, false, false);
        }
    }

    // Scatter: lane owns output column n = m for 8 edge-rows. Unconditional:
    // invalid edges contributed zero rows, so the adds are value no-ops.
#pragma unroll
    for (int ridx = 0; ridx < 8; ++ridx) {
        const int em = ridx + hi * 8;           // edge slot within tile
        const int cm = __shfl(c, em, 32);       // that edge's target node
        atomicAdd(&aggr[cm * 16 + m], acc[ridx]);
    }
}

// ---------------------------------------------------------------------------
// h1 = relu( x @ Wc1 + bc1 + aggr / max(cnt,1) )
// ---------------------------------------------------------------------------
__global__ void combine1_kernel(const float* __restrict__ x,
                                const float* __restrict__ Wc, const float* __restrict__ bc,
                                const float* __restrict__ aggr, const float* __restrict__ cnt,
                                float* __restrict__ h1, int N) {
    int i = blockIdx.x * blockDim.x + threadIdx.x;
    if (i >= N) return;
    const float x0 = x[i * 3 + 0], x1 = x[i * 3 + 1], x2 = x[i * 3 + 2];
    const float inv = 1.0f / fmaxf(cnt[i], 1.0f);
#pragma unroll
    for (int f = 0; f < 16; ++f) {
        float v = bc[f] + x0 * Wc[0 * 16 + f] + x1 * Wc[1 * 16 + f] + x2 * Wc[2 * 16 + f]
                + aggr[i * 16 + f] * inv;
        h1[i * 16 + f] = fmaxf(v, 0.0f);
    }
}

// ---------------------------------------------------------------------------
// h2 = relu( h1 @ Wc2 + bc2 + aggr / max(cnt,1) ); pool via LDS staging
// (batch is sorted, so each 256-node block touches very few graphs -> the
//  LDS pre-reduction collapses ~1.7M contended global atomics to ~tens/block)
// ---------------------------------------------------------------------------
#define G_MAX 64
__global__ void combine2_pool_kernel(const float* __restrict__ h1,
                                     const float* __restrict__ Wc, const float* __restrict__ bc,
                                     const float* __restrict__ aggr, const float* __restrict__ cnt,
                                     const int* __restrict__ batch,
                                     float* __restrict__ gsum, float* __restrict__ gcnt, int N) {
    __shared__ float lsum[G_MAX * 16];
    __shared__ float lcnt[G_MAX];
    for (int t = threadIdx.x; t < G_MAX * 16; t += blockDim.x) lsum[t] = 0.0f;
    for (int t = threadIdx.x; t < G_MAX; t += blockDim.x) lcnt[t] = 0.0f;
    __syncthreads();

    const int i = blockIdx.x * blockDim.x + threadIdx.x;
    if (i < N) {
        float h[16];
        const float4* hp = (const float4*)(h1 + i * 16);
#pragma unroll
        for (int q = 0; q < 4; ++q) {
            float4 v = hp[q];
            h[q * 4 + 0] = v.x; h[q * 4 + 1] = v.y; h[q * 4 + 2] = v.z; h[q * 4 + 3] = v.w;
        }
        const float inv = 1.0f / fmaxf(cnt[i], 1.0f);
        const int   b   = batch[i];
#pragma unroll
        for (int f = 0; f < 16; ++f) {
            float v = bc[f];
#pragma unroll
            for (int k = 0; k < 16; ++k) v += h[k] * Wc[k * 16 + f];
            v += aggr[i * 16 + f] * inv;
            atomicAdd(&lsum[b * 16 + f], fmaxf(v, 0.0f));
        }
        atomicAdd(&lcnt[b], 1.0f);
    }
    __syncthreads();

    for (int t = threadIdx.x; t < G_MAX * 16; t += blockDim.x) {
        float v = lsum[t];
        if (v != 0.0f) atomicAdd(&gsum[t], v);
    }
    for (int t = threadIdx.x; t < G_MAX; t += blockDim.x) {
        float v = lcnt[t];
        if (v != 0.0f) atomicAdd(&gcnt[t], v);
    }
}

// ---------------------------------------------------------------------------
// Head: g = gsum/max(gcnt,1); out = relu(g@Wl1+bl1) @ Wl2 + bl2
// ---------------------------------------------------------------------------
__global__ void head_kernel(const float* __restrict__ gsum, const float* __restrict__ gcnt,
                            const float* __restrict__ Wl1, const float* __restrict__ bl1,
                            const float* __restrict__ Wl2, const float* __restrict__ bl2,
                            float* __restrict__ out, int G) {
    int g = blockIdx.x * blockDim.x + threadIdx.x;
    if (g >= G) return;
    const float inv = 1.0f / fmaxf(gcnt[g], 1.0f);
    float gv[16];
#pragma unroll
    for (int k = 0; k < 16; ++k) gv[k] = gsum[g * 16 + k] * inv;
    float t[16];
#pragma unroll
    for (int f = 0; f < 16; ++f) {
        float v = bl1[f];
#pragma unroll
        for (int k = 0; k < 16; ++k) v += gv[k] * Wl1[k * 16 + f];
        t[f] = fmaxf(v, 0.0f);
    }
#pragma unroll
    for (int o = 0; o < 2; ++o) {
        float v = bl2[o];
#pragma unroll
        for (int f = 0; f < 16; ++f) v += t[f] * Wl2[f * 2 + o];
        out[g * 2 + o] = v;
    }
}

// ---------------------------------------------------------------------------
extern "C" void kernel_launch(void* const* d_in, const int* in_sizes, int n_in,
                              void* d_out, int out_size, void* d_ws, size_t ws_size,
                              hipStream_t stream) {
    const float* x     = (const float*)d_in[0];
    const int*   ei    = (const int*)  d_in[1];
    const float* eattr = (const float*)d_in[2];
    const int*   batch = (const int*)  d_in[3];
    const float* Wc1   = (const float*)d_in[4];
    const float* bc1   = (const float*)d_in[5];
    const float* Wn1   = (const float*)d_in[6];
    const float* Wc2   = (const float*)d_in[7];
    const float* bc2   = (const float*)d_in[8];
    const float* Wn2   = (const float*)d_in[9];
    const float* Wl1   = (const float*)d_in[10];
    const float* bl1   = (const float*)d_in[11];
    const float* Wl2   = (const float*)d_in[12];
    const float* bl2   = (const float*)d_in[13];

    const int N = in_sizes[0] / 3;
    const int E = in_sizes[1] / 2;
    const int G = out_size / 2;

    // workspace layout (floats)
    float* ws   = (float*)d_ws;
    float* deg  = ws;                         // [N]
    float* cnt  = deg  + (size_t)N;           // [N]
    float* dis  = cnt  + (size_t)N;           // [N]
    float* aggr = dis  + (size_t)N;           // [N*16] (reused by both layers)
    float* h1   = aggr + (size_t)N * 16;      // [N*16]
    float* gsum = h1   + (size_t)N * 16;      // [G*16]
    float* gcnt = gsum + (size_t)G * 16;      // [G]
    (void)ws_size; (void)n_in;

    hipMemsetAsync(deg,  0, sizeof(float) * (size_t)N * 2, stream);       // deg+cnt
    hipMemsetAsync(gsum, 0, sizeof(float) * (size_t)(G * 16 + G), stream);
    hipMemsetAsync(aggr, 0, sizeof(float) * (size_t)N * 16, stream);

    const int threads = 256;
    degree_kernel<<<(E + threads - 1) / threads, threads, 0, stream>>>(ei, deg, cnt, E);
    dis_kernel<<<(N + threads - 1) / threads, threads, 0, stream>>>(deg, dis, N);

    const int tiles  = (E + 15) / 16;
    const int wpb    = threads / 32;                 // 8 waves (tiles) per block
    const int eblks  = (tiles + wpb - 1) / wpb;

    // layer 1 edges: F_IN=3, K=5 -> 2 WMMA(16x16x4_f32) steps
    edge_msg_wmma<3><<<eblks, threads, 0, stream>>>(ei, x, eattr, dis, Wn1, aggr, E);
    combine1_kernel<<<(N + threads - 1) / threads, threads, 0, stream>>>(x, Wc1, bc1, aggr, cnt, h1, N);

    // layer 2 edges: F_IN=16, K=18 -> 5 WMMA steps
    hipMemsetAsync(aggr, 0, sizeof(float) * (size_t)N * 16, stream);
    edge_msg_wmma<16><<<eblks, threads, 0, stream>>>(ei, h1, eattr, dis, Wn2, aggr, E);
    combine2_pool_kernel<<<(N + threads - 1) / threads, threads, 0, stream>>>(
        h1, Wc2, bc2, aggr, cnt, batch, gsum, gcnt, N);

    head_kernel<<<(G + 63) / 64, 64, 0, stream>>>(gsum, gcnt, Wl1, bl1, Wl2, bl2,
                                                  (float*)d_out, G);
}